// DecoderSmoothedMaxPoolingLoss_16509854286466
// MI455X (gfx1250) — compile-verified
//
#include <hip/hip_runtime.h>
#include <hip/hip_bf16.h>
#include <math.h>

typedef __attribute__((ext_vector_type(2))) float v2f;
typedef __attribute__((ext_vector_type(8))) float v8f;

#define KDIM     100
#define WIN      60
#define OFFSET_D 40
#define TRUNC    21
#define GHALF    (TRUNC / 2)     // 10
#define SIGMA    9.0f
#define EPSV     1e-8f
#define TCHUNK   250             // frames per block chunk (4000/250 = 16 chunks)

// -log(clip(1-x, EPS, 1))
__device__ __forceinline__ float neg_log_term(float x) {
    float q = 1.0f - x;
    q = fminf(fmaxf(q, EPSV), 1.0f);
    return -__logf(q);
}

// Block-wide sum for 256-thread blocks (8 wave32s).
// Per-wave stage uses V_WMMA_F32_16X16X4_F32 with A = ones(16x4):
// D[m,n] = sum_k B[k,n]; summing row 0 over n gives the sum of all 64 B slots.
// Each lane contributes its value in B vgpr0 (vgpr1 = 0) -> layout-agnostic total.
// EXEC is all ones here: every wave executes this unconditionally.
template <int NWAVES>
__device__ float block_reduce_wmma(float v) {
    __shared__ float wsum[NWAVES];
    v2f a; a.x = 1.0f; a.y = 1.0f;
    v2f b; b.x = v;    b.y = 0.0f;
    v8f c = {0.f, 0.f, 0.f, 0.f, 0.f, 0.f, 0.f, 0.f};
    c = __builtin_amdgcn_wmma_f32_16x16x4_f32(false, a, false, b,
                                              (short)0, c, false, false);
    float r = c[0];                 // lanes 0..15 hold column sums (row M=0)
    r += __shfl_xor(r, 8, 32);      // reduce the 16 column sums (stays in lane group)
    r += __shfl_xor(r, 4, 32);
    r += __shfl_xor(r, 2, 32);
    r += __shfl_xor(r, 1, 32);
    const int lane = threadIdx.x & 31;
    const int wid  = threadIdx.x >> 5;
    if (lane == 0) wsum[wid] = r;
    __syncthreads();
    float tot = 0.0f;
    if (threadIdx.x == 0) {
#pragma unroll
        for (int i = 0; i < NWAVES; ++i) tot += wsum[i];   // fixed order
    }
    return tot;                    // valid on thread 0
}

// ---- Kernel 1: streaming sum of -log(1-X) over valid frames -----------------
// Length mask is a prefix in t, so each (batch, chunk) block reduces a fully
// contiguous region: no per-element masking, pure b128 streaming loads.
__global__ __launch_bounds__(256) void k_neg(const float* __restrict__ X,
                                             const int* __restrict__ lengths,
                                             float* __restrict__ partials,
                                             int T, int nchunk) {
    const int b  = blockIdx.y;
    const int c0 = blockIdx.x * TCHUNK;
    int tv = lengths[b] - c0;
    tv = tv < 0 ? 0 : (tv > TCHUNK ? TCHUNK : tv);
    const long base = ((long)b * T + c0) * KDIM;       // 400B-aligned -> 16B OK
    const float4* __restrict__ p = (const float4*)(X + base);
    const int n4 = tv * (KDIM / 4);                    // exact: 25 float4/frame
    float s = 0.0f;
    for (int i = threadIdx.x; i < n4; i += 256) {
        float4 v = p[i];
        s += neg_log_term(v.x) + neg_log_term(v.y) +
             neg_log_term(v.z) + neg_log_term(v.w);
    }
    const float tot = block_reduce_wmma<8>(s);
    if (threadIdx.x == 0) partials[b * nchunk + blockIdx.x] = tot;
}

// ---- Kernel 2: per-batch window correction + smoothed-max positive term -----
__global__ __launch_bounds__(64) void k_pos(const float* __restrict__ X,
                                            const int* __restrict__ lengths,
                                            const int* __restrict__ tgt,
                                            const int* __restrict__ w_end,
                                            float* __restrict__ out, int T) {
    const int b   = blockIdx.x;
    const int len = lengths[b];
    const int kk  = tgt[b];
    int tau_s = w_end[b] + OFFSET_D - WIN;
    tau_s = tau_s > 0 ? tau_s : 0;
    int start = tau_s;                                  // dynamic_slice clamp
    if (start > T - WIN) start = T - WIN;
    if (start < 0) start = 0;
    int tau_e = tau_s + WIN;
    if (tau_e > len) tau_e = len;
    int Lw = tau_e - tau_s;
    Lw = Lw < 0 ? 0 : (Lw > WIN ? WIN : Lw);

    __shared__ float win[WIN];
    __shared__ float corrbuf[64];
    __shared__ float smax[64];
    const int i = threadIdx.x;

    float corr = 0.0f;
    if (i < WIN) {
        float wv = 0.0f;
        if (i < Lw) {
            const float x = X[((long)b * T + start + i) * KDIM + kk];
            wv = x;
            float q = 1.0f - x;
            q = fminf(fmaxf(q, EPSV), 1.0f);
            corr = -logf(q);
        }
        win[i] = wv;                                    // zero-padded ragged win
    }
    corrbuf[i] = corr;
    __syncthreads();

    // Truncated, normalized Gaussian taps, as in reference (float32 throughout)
    float g[TRUNC];
    float gs = 0.0f;
#pragma unroll
    for (int m = 0; m < TRUNC; ++m) {
        const float xx = (float)(m - GHALF) / SIGMA;
        const float e  = expf(-0.5f * xx * xx);
        g[m] = e;
        gs  += e;
    }
#pragma unroll
    for (int m = 0; m < TRUNC; ++m) g[m] /= gs;

    // 'same' conv with pads (29,30): smoothed[o] = sum_m win[o + m - 9] * g[m]
    float sm = 0.0f;
    if (i < Lw) {
        float acc = 0.0f;
#pragma unroll
        for (int m = 0; m < TRUNC; ++m) {
            const int j = i + m - (GHALF - 1);
            const float wv = (j >= 0 && j < WIN) ? win[j] : 0.0f;
            acc += wv * g[m];
        }
        acc = fminf(fmaxf(acc, EPSV), 1.0f);
        sm = acc;                                       // invalid slots stay 0
    }
    smax[i] = sm;
    __syncthreads();

    if (i == 0) {
        float mx = 0.0f, cs = 0.0f;
        for (int j = 0; j < 64; ++j) {                  // fixed order
            mx = fmaxf(mx, smax[j]);
            cs += corrbuf[j];
        }
        out[b] = -logf(mx) - cs;                        // pos term minus window correction
    }
}

// ---- Kernel 3: deterministic final reduction of all partials ----------------
__global__ __launch_bounds__(256) void k_final(const float* __restrict__ ws,
                                               float* __restrict__ out, int n) {
    float s = 0.0f;
    for (int i = threadIdx.x; i < n; i += 256) s += ws[i];
    const float tot = block_reduce_wmma<8>(s);
    if (threadIdx.x == 0) out[0] = tot;
}

extern "C" void kernel_launch(void* const* d_in, const int* in_sizes, int n_in,
                              void* d_out, int out_size, void* d_ws, size_t ws_size,
                              hipStream_t stream) {
    const float* X       = (const float*)d_in[0];
    const int*   lengths = (const int*)d_in[1];
    const int*   tgt     = (const int*)d_in[2];
    const int*   w_end   = (const int*)d_in[3];

    const int  B  = in_sizes[1];
    const long NE = (long)in_sizes[0];
    const int  T  = (int)(NE / ((long)B * KDIM));
    const int  nchunk = (T + TCHUNK - 1) / TCHUNK;

    float* partials  = (float*)d_ws;                    // B*nchunk block sums
    float* batchvals = partials + (size_t)B * nchunk;   // B per-batch terms

    k_neg  <<<dim3(nchunk, B), 256, 0, stream>>>(X, lengths, partials, T, nchunk);
    k_pos  <<<dim3(B),          64, 0, stream>>>(X, lengths, tgt, w_end, batchvals, T);
    k_final<<<1,               256, 0, stream>>>(partials, (float*)d_out,
                                                 B * nchunk + B);
}